// BiAlignLayer_21818433864131
// MI455X (gfx1250) — compile-verified
//
#include <hip/hip_runtime.h>
#include <hip/hip_bf16.h>

typedef __attribute__((ext_vector_type(2))) float v2f;
typedef __attribute__((ext_vector_type(4))) float v4f;
typedef __attribute__((ext_vector_type(8))) float v8f;

// Problem constants from the reference: B=32, L=1024, D=512, NN=512
#define BB   32
#define LL   1024
#define DD   512
#define NN   512
#define CH   64          // L-chunks for the streaming reduction
#define LC   (LL / CH)   // 16 rows per chunk

// ---------------------------------------------------------------------------
// Kernel 1: partial[b][ch][d] = sum over 16 l's of (i[b,l,d] - j[b,l,d]).
// Grid (32, 64) x 128 threads; each thread owns one float4 column (d4) and
// streams 16 rows of i and j with fully coalesced 128-bit loads.
// 128 MB read total -> HBM-bound stage (~5.5 us at 23.3 TB/s).
// ---------------------------------------------------------------------------
__global__ __launch_bounds__(128) void bialign_diff_partial(
    const float* __restrict__ ip, const float* __restrict__ jp,
    float* __restrict__ part) {
  const int b  = blockIdx.x;
  const int ch = blockIdx.y;
  const int d4 = threadIdx.x;           // 0..127 -> d = 4*d4
  const size_t base = (size_t)b * LL * DD + (size_t)ch * LC * DD;
  const v4f* __restrict__ iv = (const v4f*)(ip + base) + d4;
  const v4f* __restrict__ jv = (const v4f*)(jp + base) + d4;
  v4f acc = (v4f)0.0f;
#pragma unroll
  for (int l = 0; l < LC; ++l) {
    acc += iv[l * (DD / 4)] - jv[l * (DD / 4)];
  }
  v4f* pv = (v4f*)(part + ((size_t)(b * CH + ch)) * DD) + d4;
  *pv = acc;
}

// ---------------------------------------------------------------------------
// Kernel 2: x[b][d] = (1/1024) * sum_ch partial[b][ch][d]   (16384 threads)
// ---------------------------------------------------------------------------
__global__ __launch_bounds__(256) void bialign_final_reduce(
    const float* __restrict__ part, float* __restrict__ x) {
  const int idx = blockIdx.x * 256 + threadIdx.x;  // 0..16383
  const int b = idx >> 9;
  const int d = idx & (DD - 1);
  float s = 0.0f;
#pragma unroll 8
  for (int c = 0; c < CH; ++c) {
    s += part[((size_t)(b * CH + c)) * DD + d];
  }
  x[idx] = s * (1.0f / (float)LL);
}

// ---------------------------------------------------------------------------
// Kernel 3: y = x @ W_agg (M=32, N=512, K=512) via V_WMMA_F32_16X16X4_F32,
// fused epilogue: out = 0.5*(relu(y + bias) + relu(-y + bias)).
// One wave32 per 16x16 output tile; K=512 -> 128 chained WMMA accumulations.
// fp32 WMMA keeps full precision vs the fp32 reference.
//
// A (16x4 f32): lane = m + 16*(k>>1), vgpr = k&1
// B (4x16 f32): lane = n + 16*(k>>1), vgpr = k&1
// C/D (16x16 f32): n = lane&15, m = reg + 8*(lane>=16)
// ---------------------------------------------------------------------------
__global__ __launch_bounds__(32) void bialign_wmma_gemm(
    const float* __restrict__ x,     // [32, 512]
    const float* __restrict__ W,     // [512, 512] row-major (k, n)
    const float* __restrict__ bias,  // [512]
    float* __restrict__ out) {       // [32, 512]
  const int nt   = blockIdx.x;        // 0..31  (N tiles)
  const int mt   = blockIdx.y;        // 0..1   (M tiles)
  const int lane = threadIdx.x;       // wave32
  const int half = lane >> 4;         // 0: K pair {0,1}; 1: K pair {2,3}
  const int lm   = lane & 15;

  const int mrow = mt * 16 + lm;      // row of x this lane loads for A
  const int ncol = nt * 16 + lm;      // column of W this lane loads for B

  v8f c = {};
#pragma unroll 4
  for (int k = 0; k < DD; k += 4) {
    const int kb = k + half * 2;
    v2f a, b;
    a.x = x[(size_t)mrow * DD + kb];
    a.y = x[(size_t)mrow * DD + kb + 1];
    b.x = W[(size_t)kb * NN + ncol];
    b.y = W[(size_t)(kb + 1) * NN + ncol];
    // D = A*B + C ; 8 args: (neg_a, A, neg_b, B, c_mod, C, reuse_a, reuse_b)
    c = __builtin_amdgcn_wmma_f32_16x16x4_f32(
        false, a, false, b, (short)0, c, false, false);
  }

  const float bb = bias[ncol];
#pragma unroll
  for (int r = 0; r < 8; ++r) {
    const int m = mt * 16 + r + half * 8;
    const float s = c[r];
    // 0.5*(relu(xW + b) + relu(-xW + b))
    const float v = 0.5f * (fmaxf(s + bb, 0.0f) + fmaxf(bb - s, 0.0f));
    out[(size_t)m * NN + ncol] = v;
  }
}

// ---------------------------------------------------------------------------
// Launch: inputs are (i, j, W_agg, b_agg) fp32; output [32, 512] fp32.
// Workspace: partial sums (4 MB) + reduced x (64 KB).
// ---------------------------------------------------------------------------
extern "C" void kernel_launch(void* const* d_in, const int* in_sizes, int n_in,
                              void* d_out, int out_size, void* d_ws, size_t ws_size,
                              hipStream_t stream) {
  const float* i_in = (const float*)d_in[0];
  const float* j_in = (const float*)d_in[1];
  const float* W    = (const float*)d_in[2];
  const float* bias = (const float*)d_in[3];
  float* out = (float*)d_out;

  float* part = (float*)d_ws;                       // [32][64][512] = 4 MB
  float* x    = part + (size_t)BB * CH * DD;        // [32][512]     = 64 KB

  dim3 g1(BB, CH);
  bialign_diff_partial<<<g1, 128, 0, stream>>>(i_in, j_in, part);

  bialign_final_reduce<<<(BB * DD) / 256, 256, 0, stream>>>(part, x);

  dim3 g3(NN / 16, BB / 16);
  bialign_wmma_gemm<<<g3, 32, 0, stream>>>(x, W, bias, out);
}